// YOWOF_7902739825060
// MI455X (gfx1250) — compile-verified
//
#include <hip/hip_runtime.h>
#include <hip/hip_bf16.h>

// YOWOF-style detection head for gfx1250 (MI455X).
// Phase 1 (memory-bound, 189MB stream): async global->LDS (b128) pipelined
//   channel max/argmax. prob = sigmoid(cls+obj-log(1+e^cls'+e^obj')) is
//   strictly monotone in cls for fixed obj, so argmax over prob == argmax
//   over raw cls; only one sigmoid per anchor is needed.
// Phase 2: 16-bit-prefix radix select + exact bitonic sort of candidates.
// Phase 3: box decode + greedy class-aware NMS in one workgroup.

#define HW_      65536     // 256*256
#define IMG_     2048
#define NA_      9
#define NC_      80
#define NPOS_    589824    // HW_*NA_
#define TOPK_    1000
#define CONF_    0.05f
#define NMS_T_   0.6f
#define CTR_CLAMP_   32.0f
#define EXP_CLAMP_   18.420680743952367f   // log(1e8)
#define SCALE_CLAMP_ 4.135166556742356f    // log(1000/16)
#define CAP_     4096
#define DEPTH_   4

// ---------------- Phase 1: per-anchor score + label --------------------
// grid = (HW_/1024, NA_); block = 256 threads; each thread owns 4 positions.
__global__ __launch_bounds__(256) void k_scores(const float* __restrict__ obj,
                                                const float* __restrict__ cls,
                                                float* __restrict__ scores,
                                                int* __restrict__ labels) {
  __shared__ float buf[DEPTH_][1024];
  const int t    = threadIdx.x;
  const int a    = blockIdx.y;                   // anchor group (independent)
  const int p4   = blockIdx.x * 1024 + t * 4;    // first of this thread's 4 positions
  const unsigned vaddr = (unsigned)(p4 * 4);     // per-lane byte offset in a channel row
  const float* clsa = cls + (unsigned long long)a * NC_ * HW_;

  // Prologue: issue DEPTH_ async b128 channel-row loads (channels 0..3).
  for (int p = 0; p < DEPTH_; ++p) {
    unsigned lds = (unsigned)(unsigned long long)&buf[p][t * 4];
    unsigned long long sa =
        (unsigned long long)(const void*)(clsa + (unsigned long long)p * HW_);
    asm volatile("global_load_async_to_lds_b128 %0, %1, %2"
                 :: "v"(lds), "v"(vaddr), "s"(sa) : "memory");
  }

  float m0 = -1e30f, m1 = -1e30f, m2 = -1e30f, m3 = -1e30f;
  int   l0 = 0,      l1 = 0,      l2 = 0,      l3 = 0;

  for (int c = 0; c < NC_; ++c) {
    // Wait until the oldest outstanding async load (channel c) landed.
    int rem = (NC_ - 1) - c;                     // remaining in-flight after oldest
    if (rem >= 3)      asm volatile("s_wait_asynccnt 3" ::: "memory");
    else if (rem == 2) asm volatile("s_wait_asynccnt 2" ::: "memory");
    else if (rem == 1) asm volatile("s_wait_asynccnt 1" ::: "memory");
    else               asm volatile("s_wait_asynccnt 0" ::: "memory");

    const float4 v = *(const float4*)&buf[c & (DEPTH_ - 1)][t * 4]; // ds_load_b128
    if (v.x > m0) { m0 = v.x; l0 = c; }          // strict > keeps first max (top_k tie rule)
    if (v.y > m1) { m1 = v.y; l1 = c; }
    if (v.z > m2) { m2 = v.z; l2 = c; }
    if (v.w > m3) { m3 = v.w; l3 = c; }

    // Ensure the LDS read retired before the ring slot is rewritten.
    asm volatile("s_wait_dscnt 0" ::: "memory");
    int nc = c + DEPTH_;
    if (nc < NC_) {
      unsigned lds = (unsigned)(unsigned long long)&buf[nc & (DEPTH_ - 1)][t * 4];
      unsigned long long sa =
          (unsigned long long)(const void*)(clsa + (unsigned long long)nc * HW_);
      asm volatile("global_load_async_to_lds_b128 %0, %1, %2"
                   :: "v"(lds), "v"(vaddr), "s"(sa) : "memory");
    }
  }

  const float4 ov = *(const float4*)(obj + (unsigned long long)a * HW_ + p4);
  float mm[4] = {m0, m1, m2, m3};
  int   ll[4] = {l0, l1, l2, l3};
  float oo[4] = {ov.x, ov.y, ov.z, ov.w};
#pragma unroll
  for (int i = 0; i < 4; ++i) {
    float m = mm[i], o = oo[i];
    float L  = log1pf(__expf(fminf(m, EXP_CLAMP_)) + __expf(fminf(o, EXP_CLAMP_)));
    float nm = m + o - L;
    float prob = 1.0f / (1.0f + __expf(-nm));
    scores[(unsigned long long)(p4 + i) * NA_ + a] = prob;
    labels[(unsigned long long)(p4 + i) * NA_ + a] = ll[i];
  }
}

// ---------------- Phase 2: radix-select top-k -------------------------
__global__ __launch_bounds__(256) void k_init(unsigned* hist, int* cnt) {
  int i = blockIdx.x * 256 + threadIdx.x;
  if (i < 65536) hist[i] = 0u;
  if (i == 65536) *cnt = 0;
}

__global__ __launch_bounds__(256) void k_hist(const float* __restrict__ scores,
                                              unsigned* __restrict__ hist) {
  int i = blockIdx.x * 256 + threadIdx.x;        // exactly NPOS_ threads
  unsigned b = __float_as_uint(scores[i]) >> 16; // scores in (0,1] -> bins <= 0x3F80
  atomicAdd(&hist[b], 1u);
}

__global__ __launch_bounds__(256) void k_scan(const unsigned* __restrict__ hist,
                                              int* __restrict__ sel) {
  __shared__ unsigned csum[256];
  unsigned s = 0;
  for (int b = 0; b < 256; ++b) s += hist[threadIdx.x * 256 + b];
  csum[threadIdx.x] = s;
  __syncthreads();
  if (threadIdx.x == 0) {
    unsigned acc = 0;
    for (int c = 255; c >= 0; --c) {
      if (acc + csum[c] >= (unsigned)TOPK_) {
        unsigned a2 = acc;
        for (int b = 255; b >= 0; --b) {
          unsigned h = hist[c * 256 + b];
          if (a2 + h >= (unsigned)TOPK_) { sel[0] = c * 256 + b; sel[1] = (int)a2; return; }
          a2 += h;
        }
      }
      acc += csum[c];
    }
    sel[0] = 0; sel[1] = 0;
  }
}

__global__ __launch_bounds__(256) void k_collect(const float* __restrict__ scores,
                                                 const int* __restrict__ sel,
                                                 unsigned long long* __restrict__ cand,
                                                 int* __restrict__ cnt) {
  int i = blockIdx.x * 256 + threadIdx.x;
  unsigned bits = __float_as_uint(scores[i]);
  if ((int)(bits >> 16) >= sel[0]) {
    int p = atomicAdd(cnt, 1);
    if (p < CAP_) {
      // key: descending score, then ascending index (matches lax.top_k)
      cand[p] = ((unsigned long long)bits << 32) | (unsigned)(~(unsigned)i);
    }
  }
}

__global__ __launch_bounds__(1024) void k_sortsel(const unsigned long long* __restrict__ cand,
                                                  const int* __restrict__ cnt,
                                                  unsigned long long* __restrict__ topk) {
  __shared__ unsigned long long s[CAP_];
  int n = *cnt; if (n > CAP_) n = CAP_;
  for (int i = threadIdx.x; i < CAP_; i += 1024) s[i] = (i < n) ? cand[i] : 0ULL;
  __syncthreads();
  for (int k = 2; k <= CAP_; k <<= 1) {
    for (int j = k >> 1; j > 0; j >>= 1) {
      for (int e = 0; e < CAP_ / 1024; ++e) {
        int i = threadIdx.x + e * 1024;
        int ixj = i ^ j;
        if (ixj > i) {
          bool up = (i & k) == 0;                 // descending overall
          unsigned long long A = s[i], B = s[ixj];
          bool sw = up ? (A < B) : (A > B);
          if (sw) { s[i] = B; s[ixj] = A; }
        }
      }
      __syncthreads();
    }
  }
  for (int i = threadIdx.x; i < TOPK_; i += 1024) topk[i] = s[i];
}

// ---------------- Phase 3: decode + NMS -------------------------------
__global__ __launch_bounds__(256) void k_decode(const unsigned long long* __restrict__ topk,
                                                const int* __restrict__ labels_all,
                                                const float* __restrict__ reg,
                                                const float* __restrict__ anc,
                                                float* __restrict__ out) {
  int i = blockIdx.x * 256 + threadIdx.x;
  if (i >= TOPK_) return;
  unsigned long long key = topk[i];
  float score  = __uint_as_float((unsigned)(key >> 32));
  unsigned idx = ~(unsigned)key;
  int lbl = labels_all[idx];
  unsigned pos = idx / NA_, a = idx % NA_;
  float r0 = reg[(a * 4 + 0) * HW_ + pos];
  float r1 = reg[(a * 4 + 1) * HW_ + pos];
  float r2 = reg[(a * 4 + 2) * HW_ + pos];
  float r3 = reg[(a * 4 + 3) * HW_ + pos];
  float ax = anc[idx * 4 + 0], ay = anc[idx * 4 + 1];
  float aw = anc[idx * 4 + 2], ah = anc[idx * 4 + 3];
  float ox = fminf(fmaxf(r0 * aw, -CTR_CLAMP_), CTR_CLAMP_);
  float oy = fminf(fmaxf(r1 * ah, -CTR_CLAMP_), CTR_CLAMP_);
  float cx = ax + ox, cy = ay + oy;
  float w = aw * __expf(fminf(r2, SCALE_CLAMP_));
  float h = ah * __expf(fminf(r3, SCALE_CLAMP_));
  const float inv = 1.0f / (float)IMG_;
  float x1 = fminf(fmaxf((cx - 0.5f * w) * inv, 0.f), 1.f);
  float y1 = fminf(fmaxf((cy - 0.5f * h) * inv, 0.f), 1.f);
  float x2 = fminf(fmaxf((cx + 0.5f * w) * inv, 0.f), 1.f);
  float y2 = fminf(fmaxf((cy + 0.5f * h) * inv, 0.f), 1.f);
  out[i]                     = score;
  out[TOPK_ + i]             = (float)lbl;
  out[2 * TOPK_ + i * 4 + 0] = x1;
  out[2 * TOPK_ + i * 4 + 1] = y1;
  out[2 * TOPK_ + i * 4 + 2] = x2;
  out[2 * TOPK_ + i * 4 + 3] = y2;
}

__global__ __launch_bounds__(1024) void k_nms(float* __restrict__ out) {
  __shared__ float x1s[TOPK_], y1s[TOPK_], x2s[TOPK_], y2s[TOPK_], ar[TOPK_];
  __shared__ int lb[TOPK_];
  __shared__ int kp[TOPK_];
  for (int i = threadIdx.x; i < TOPK_; i += 1024) {
    float a1 = out[2 * TOPK_ + i * 4 + 0], b1 = out[2 * TOPK_ + i * 4 + 1];
    float a2 = out[2 * TOPK_ + i * 4 + 2], b2 = out[2 * TOPK_ + i * 4 + 3];
    x1s[i] = a1; y1s[i] = b1; x2s[i] = a2; y2s[i] = b2;
    ar[i] = (a2 - a1) * (b2 - b1);
    lb[i] = (int)out[TOPK_ + i];
    kp[i] = (out[i] >= CONF_) ? 1 : 0;
  }
  for (int i = 0; i < TOPK_; ++i) {
    __syncthreads();
    if (!kp[i]) continue;
    float X1 = x1s[i], Y1 = y1s[i], X2 = x2s[i], Y2 = y2s[i], A = ar[i];
    int L = lb[i];
    for (int j = i + 1 + threadIdx.x; j < TOPK_; j += 1024) {
      if (kp[j] && lb[j] == L) {
        float xx1 = fmaxf(X1, x1s[j]), yy1 = fmaxf(Y1, y1s[j]);
        float xx2 = fminf(X2, x2s[j]), yy2 = fminf(Y2, y2s[j]);
        float inter = fmaxf(1e-28f, xx2 - xx1) * fmaxf(1e-28f, yy2 - yy1);
        float iou = inter / (A + ar[j] - inter + 1e-14f);
        if (iou > NMS_T_) kp[j] = 0;
      }
    }
  }
  __syncthreads();
  for (int i = threadIdx.x; i < TOPK_; i += 1024) out[6 * TOPK_ + i] = kp[i] ? 1.0f : 0.0f;
}

extern "C" void kernel_launch(void* const* d_in, const int* in_sizes, int n_in,
                              void* d_out, int out_size, void* d_ws, size_t ws_size,
                              hipStream_t stream) {
  (void)in_sizes; (void)n_in; (void)out_size; (void)ws_size;
  const float* obj = (const float*)d_in[0];
  const float* cls = (const float*)d_in[1];
  const float* reg = (const float*)d_in[2];
  const float* anc = (const float*)d_in[3];
  float* out = (float*)d_out;
  char* ws = (char*)d_ws;

  float*              scores_all = (float*)(ws + 0);         // NPOS_ floats
  int*                labels_all = (int*)(ws + 2359296);     // NPOS_ ints
  unsigned*           hist       = (unsigned*)(ws + 4718592);// 65536 u32
  int*                sel        = (int*)(ws + 4980736);     // 2 ints
  int*                cnt        = (int*)(ws + 4980992);     // 1 int
  unsigned long long* cand       = (unsigned long long*)(ws + 4981248); // CAP_ u64
  unsigned long long* topk       = (unsigned long long*)(ws + 5014016); // TOPK_ u64

  k_init   <<<257,              256, 0, stream>>>(hist, cnt);
  k_scores <<<dim3(HW_ / 1024, NA_), 256, 0, stream>>>(obj, cls, scores_all, labels_all);
  k_hist   <<<NPOS_ / 256,      256, 0, stream>>>(scores_all, hist);
  k_scan   <<<1,                256, 0, stream>>>(hist, sel);
  k_collect<<<NPOS_ / 256,      256, 0, stream>>>(scores_all, sel, cand, cnt);
  k_sortsel<<<1,               1024, 0, stream>>>(cand, cnt, topk);
  k_decode <<<4,                256, 0, stream>>>(topk, labels_all, reg, anc, out);
  k_nms    <<<1,               1024, 0, stream>>>(out);
}